// LSTMModel_CenterlineEmbed_59382217834743
// MI455X (gfx1250) — compile-verified
//
#include <hip/hip_runtime.h>

typedef float v2f __attribute__((ext_vector_type(2)));
typedef float v8f __attribute__((ext_vector_type(8)));

#define NWAVES 8
#define HSTR   100   // h-stage row stride (floats): conflict-free for both access patterns

// ---- LDS layout (float offsets), all 16B-aligned ----
#define DECW_OFF   0        // 24 ntiles * 24 ksteps * 32 lanes * 2 = 36864  (Wih_d + Whh_d fused)
#define ENCW_OFF   36864    // 16 * 16 * 64 = 16384
#define CENTW_OFF  53248    // 8 * 8 * 64   = 4096
#define ENCX_OFF   57344    // 16 * 64      = 1024
#define CENTX_OFF  58368    // 8 * 64       = 512
#define BIASD_OFF  58880    // 384
#define BIASE_OFF  59264    // 256
#define BIASC_OFF  59520    // 128
#define WEMB_OFF   59648    // 192
#define BEMB_OFF   59840    // 2 (+pad)
#define HST_OFF    59844    // NWAVES * 16 * HSTR = 12800
#define LDS_FLOATS 72644
#define LDS_BYTES  (LDS_FLOATS * 4)

__device__ __forceinline__ v8f wmma4(v2f a, v2f b, v8f c) {
  // D = A(16x4 f32) * B(4x16 f32) + C(16x16 f32)
  return __builtin_amdgcn_wmma_f32_16x16x4_f32(false, a, false, b, (short)0, c, false, false);
}
__device__ __forceinline__ float sigm_(float x) { return 1.0f / (1.0f + __expf(-x)); }
__device__ __forceinline__ float tanh_(float x) { return 1.0f - 2.0f / (1.0f + __expf(2.0f * x)); }

// One LSTM phase for a 16-row batch tile owned by this wave.
// GRP = hidden/16 column groups; KS = hidden/4 K-steps. Gates are [i f g o] tiles j, j+GRP, j+2GRP, j+3GRP.
template<int GRP, int KS>
__device__ __forceinline__ void lstm_phase(
    const float* __restrict__ xg, int T,
    const float* __restrict__ wfrag, const float* __restrict__ xfrag,
    const float* __restrict__ bias,
    float* __restrict__ hst, int colOff,
    v8f* cT, int lane, long rowc)
{
  const int n16 = lane & 15, half = lane >> 4;
  for (int t = 0; t < T; ++t) {
    // A-fragment of padded x: A[M][0..1] = x, A[M][2..3] = 0  -> half==1 lanes hold zeros
    v2f ax = {0.f, 0.f};
    if (half == 0) {
      const float* xp = xg + (rowc * T + t) * 2;
      ax[0] = xp[0];
      ax[1] = xp[1];
    }
    v8f hN[GRP];
    #pragma unroll
    for (int j = 0; j < GRP; ++j) {
      v8f acc[4];
      #pragma unroll
      for (int g = 0; g < 4; ++g) {
        const int nt = j + g * GRP;
        const float b = bias[nt * 16 + n16];
        acc[g] = (v8f){b, b, b, b, b, b, b, b};
      }
      // x contribution (K padded to 4)
      #pragma unroll
      for (int g = 0; g < 4; ++g) {
        const int nt = j + g * GRP;
        v2f bf = *(const v2f*)&xfrag[(nt * 32 + lane) * 2];
        acc[g] = wmma4(ax, bf, acc[g]);
      }
      // h @ W^T, K-loop of 4
      #pragma unroll 4
      for (int ks = 0; ks < KS; ++ks) {
        v2f ah = *(const v2f*)&hst[n16 * HSTR + colOff + ks * 4 + 2 * half];
        #pragma unroll
        for (int g = 0; g < 4; ++g) {
          const int nt = j + g * GRP;
          v2f bf = *(const v2f*)&wfrag[((nt * KS + ks) * 32 + lane) * 2];
          acc[g] = wmma4(ah, bf, acc[g]);
        }
      }
      // elementwise LSTM cell update on C-layout tile
      v8f ci = cT[j], hv;
      #pragma unroll
      for (int e = 0; e < 8; ++e) {
        const float ig = sigm_(acc[0][e]);
        const float fg = sigm_(acc[1][e]);
        const float gg = tanh_(acc[2][e]);
        const float og = sigm_(acc[3][e]);
        const float cn = fg * ci[e] + ig * gg;
        ci[e] = cn;
        hv[e] = og * tanh_(cn);
      }
      cT[j] = ci;
      hN[j] = hv;
    }
    // publish new h AFTER all groups consumed old h (same-wave DS ops are in-order)
    #pragma unroll
    for (int j = 0; j < GRP; ++j) {
      #pragma unroll
      for (int e = 0; e < 8; ++e)
        hst[(e + 8 * half) * HSTR + colOff + j * 16 + n16] = hN[j][e];
    }
  }
}

__global__ void __launch_bounds__(256, 1)
traj_lstm_fused(
    const float* __restrict__ traj, const float* __restrict__ cent,
    const float* __restrict__ Wih_c, const float* __restrict__ Whh_c,
    const float* __restrict__ bih_c, const float* __restrict__ bhh_c,
    const float* __restrict__ Wih_e, const float* __restrict__ Whh_e,
    const float* __restrict__ bih_e, const float* __restrict__ bhh_e,
    const float* __restrict__ Wih_d, const float* __restrict__ Whh_d,
    const float* __restrict__ bih_d, const float* __restrict__ bhh_d,
    const float* __restrict__ W_emb, const float* __restrict__ b_emb,
    float* __restrict__ out, int B)
{
  extern __shared__ __align__(16) float lds[];
  const int tid = threadIdx.x;

  // ================= prepack: weights -> per-lane WMMA B-fragments in LDS =================
  // B(4x16) layout: lane l holds column n=l&15, rows kb=2*(l>>4) and kb+1.
  for (int i = tid; i < 24 * 24 * 32; i += 256) {            // decoder: Wsum = Wih_d + Whh_d, [384x96]
    const int f = i >> 5, l = i & 31;
    const int nt = f / 24, ks = f % 24;
    const int n = nt * 16 + (l & 15);
    const int kb = ks * 4 + 2 * (l >> 4);
    lds[DECW_OFF + i * 2 + 0] = Wih_d[n * 96 + kb + 0] + Whh_d[n * 96 + kb + 0];
    lds[DECW_OFF + i * 2 + 1] = Wih_d[n * 96 + kb + 1] + Whh_d[n * 96 + kb + 1];
  }
  for (int i = tid; i < 16 * 16 * 32; i += 256) {            // encoder Whh_e [256x64]
    const int f = i >> 5, l = i & 31;
    const int nt = f / 16, ks = f % 16;
    const int n = nt * 16 + (l & 15);
    const int kb = ks * 4 + 2 * (l >> 4);
    lds[ENCW_OFF + i * 2 + 0] = Whh_e[n * 64 + kb + 0];
    lds[ENCW_OFF + i * 2 + 1] = Whh_e[n * 64 + kb + 1];
  }
  for (int i = tid; i < 8 * 8 * 32; i += 256) {              // centerline Whh_c [128x32]
    const int f = i >> 5, l = i & 31;
    const int nt = f / 8, ks = f % 8;
    const int n = nt * 16 + (l & 15);
    const int kb = ks * 4 + 2 * (l >> 4);
    lds[CENTW_OFF + i * 2 + 0] = Whh_c[n * 32 + kb + 0];
    lds[CENTW_OFF + i * 2 + 1] = Whh_c[n * 32 + kb + 1];
  }
  for (int i = tid; i < 16 * 32; i += 256) {                 // Wih_e [256x2] padded to K=4
    const int nt = i >> 5, l = i & 31;
    const int n = nt * 16 + (l & 15);
    const bool lo = (l >> 4) == 0;
    lds[ENCX_OFF + i * 2 + 0] = lo ? Wih_e[n * 2 + 0] : 0.f;
    lds[ENCX_OFF + i * 2 + 1] = lo ? Wih_e[n * 2 + 1] : 0.f;
  }
  for (int i = tid; i < 8 * 32; i += 256) {                  // Wih_c [128x2] padded to K=4
    const int nt = i >> 5, l = i & 31;
    const int n = nt * 16 + (l & 15);
    const bool lo = (l >> 4) == 0;
    lds[CENTX_OFF + i * 2 + 0] = lo ? Wih_c[n * 2 + 0] : 0.f;
    lds[CENTX_OFF + i * 2 + 1] = lo ? Wih_c[n * 2 + 1] : 0.f;
  }
  for (int i = tid; i < 384; i += 256) lds[BIASD_OFF + i] = bih_d[i] + bhh_d[i];
  for (int i = tid; i < 256; i += 256) lds[BIASE_OFF + i] = bih_e[i] + bhh_e[i];
  for (int i = tid; i < 128; i += 256) lds[BIASC_OFF + i] = bih_c[i] + bhh_c[i];
  for (int i = tid; i < 192; i += 256) lds[WEMB_OFF + i] = W_emb[i];
  if (tid < 2) lds[BEMB_OFF + tid] = b_emb[tid];
  for (int i = tid; i < NWAVES * 16 * HSTR; i += 256) lds[HST_OFF + i] = 0.f;  // h starts at 0
  __syncthreads();

  // ================= per-wave recurrence over a 16-sample batch tile =================
  const int wv = tid >> 5, lane = tid & 31;
  const int n16 = lane & 15, half = lane >> 4;
  const long rowb = (long)(((long)blockIdx.x * NWAVES + wv) * 16 + n16);
  const long rowc = rowb < B ? rowb : (long)(B - 1);
  float* hst = &lds[HST_OFF + wv * (16 * HSTR)];

  // encoder LSTM (hidden 64, 20 steps) -> h cols [0,64)
  v8f cE[4];
  #pragma unroll
  for (int j = 0; j < 4; ++j) cE[j] = (v8f){0,0,0,0,0,0,0,0};
  lstm_phase<4, 16>(traj, 20, &lds[ENCW_OFF], &lds[ENCX_OFF], &lds[BIASE_OFF], hst, 0, cE, lane, rowc);

  // centerline LSTM (hidden 32, 100 steps) -> h cols [64,96)
  v8f cC[2];
  #pragma unroll
  for (int j = 0; j < 2; ++j) cC[j] = (v8f){0,0,0,0,0,0,0,0};
  lstm_phase<2, 8>(cent, 100, &lds[CENTW_OFF], &lds[CENTX_OFF], &lds[BIASC_OFF], hst, 64, cC, lane, rowc);

  // decoder LSTM (hidden 96, x == h, 30 steps). c0 = [enc_c | cent_c], h0 already staged in hst.
  v8f cD[6] = {cE[0], cE[1], cE[2], cE[3], cC[0], cC[1]};
  const float* wfragD = &lds[DECW_OFF];
  const float* biasD  = &lds[BIASD_OFF];
  for (int t = 0; t < 30; ++t) {
    v8f hN[6];
    #pragma unroll
    for (int j = 0; j < 6; ++j) {
      v8f acc[4];
      #pragma unroll
      for (int g = 0; g < 4; ++g) {
        const int nt = j + g * 6;
        const float b = biasD[nt * 16 + n16];
        acc[g] = (v8f){b, b, b, b, b, b, b, b};
      }
      #pragma unroll 4
      for (int ks = 0; ks < 24; ++ks) {
        v2f ah = *(const v2f*)&hst[n16 * HSTR + ks * 4 + 2 * half];
        #pragma unroll
        for (int g = 0; g < 4; ++g) {
          const int nt = j + g * 6;
          v2f bf = *(const v2f*)&wfragD[((nt * 24 + ks) * 32 + lane) * 2];
          acc[g] = wmma4(ah, bf, acc[g]);
        }
      }
      v8f ci = cD[j], hv;
      #pragma unroll
      for (int e = 0; e < 8; ++e) {
        const float ig = sigm_(acc[0][e]);
        const float fg = sigm_(acc[1][e]);
        const float gg = tanh_(acc[2][e]);
        const float og = sigm_(acc[3][e]);
        const float cn = fg * ci[e] + ig * gg;
        ci[e] = cn;
        hv[e] = og * tanh_(cn);
      }
      cD[j] = ci;
      hN[j] = hv;
    }
    #pragma unroll
    for (int j = 0; j < 6; ++j) {
      #pragma unroll
      for (int e = 0; e < 8; ++e)
        hst[(e + 8 * half) * HSTR + j * 16 + n16] = hN[j][e];
    }
    // pos = h_new @ W_emb^T + b_emb ; 32 lanes = 16 rows x 2 outputs
    {
      float s = 0.f;
      const float4* hr = (const float4*)&hst[n16 * HSTR];
      const float4* wr = (const float4*)&lds[WEMB_OFF + half * 96];
      #pragma unroll
      for (int k = 0; k < 24; ++k) {
        const float4 a = hr[k];
        const float4 w = wr[k];
        s += a.x * w.x + a.y * w.y + a.z * w.z + a.w * w.w;
      }
      s += lds[BEMB_OFF + half];
      if (rowb < B) out[(rowb * 30 + t) * 2 + half] = s;
    }
  }
}

extern "C" void kernel_launch(void* const* d_in, const int* in_sizes, int n_in,
                              void* d_out, int out_size, void* d_ws, size_t ws_size,
                              hipStream_t stream) {
  (void)n_in; (void)out_size; (void)d_ws; (void)ws_size;
  const float* traj  = (const float*)d_in[0];
  const float* cent  = (const float*)d_in[1];
  const float* Wih_c = (const float*)d_in[2];
  const float* Whh_c = (const float*)d_in[3];
  const float* bih_c = (const float*)d_in[4];
  const float* bhh_c = (const float*)d_in[5];
  const float* Wih_e = (const float*)d_in[6];
  const float* Whh_e = (const float*)d_in[7];
  const float* bih_e = (const float*)d_in[8];
  const float* bhh_e = (const float*)d_in[9];
  const float* Wih_d = (const float*)d_in[10];
  const float* Whh_d = (const float*)d_in[11];
  const float* bih_d = (const float*)d_in[12];
  const float* bhh_d = (const float*)d_in[13];
  const float* W_emb = (const float*)d_in[14];
  const float* b_emb = (const float*)d_in[15];
  float* out = (float*)d_out;

  const int B = in_sizes[0] / 40;                 // input_traj is [B, 20, 2]
  const int nblk = (B + NWAVES * 16 - 1) / (NWAVES * 16);

  (void)hipFuncSetAttribute(reinterpret_cast<const void*>(traj_lstm_fused),
                            hipFuncAttributeMaxDynamicSharedMemorySize, LDS_BYTES);

  traj_lstm_fused<<<nblk, 256, LDS_BYTES, stream>>>(
      traj, cent, Wih_c, Whh_c, bih_c, bhh_c,
      Wih_e, Whh_e, bih_e, bhh_e,
      Wih_d, Whh_d, bih_d, bhh_d,
      W_emb, b_emb, out, B);
}